// Match_Regression_12163347383133
// MI455X (gfx1250) — compile-verified
//
#include <hip/hip_runtime.h>
#include <hip/hip_bf16.h>
#include <math.h>

#define NUM_TOPICS 25
#define ATTN_DIM   10
#define BATCH      16384
#define NREV       200
#define SCALE_F    1.955f      /* (3.41 - (-0.5))/2 */
#define TMIN_F    (-0.5f)

typedef __attribute__((ext_vector_type(2))) float v2f;
typedef __attribute__((ext_vector_type(8))) float v8f;

// Workspace layout (floats):
//   [0    ..  896)  W1p : 32 x 28 zero-padded fused matrix
//                         rows 0-24: (WK^T WQ)/sqrt(d); row 25: Ws^T Wo; rest 0
//   [896  ..  922)  v1  : (WK^T bQ)/sqrt(d), v1[25] = 0
//   [928  ..  953)  w2  : WV^T Wo
//   [960]           C0  : Wo.(bV+bs) + bo
//   [1024 .. 1024+16384*32)  U : per-b [u[0..24], h, pad...] stride 32
#define WS_W1 0
#define WS_V1 896
#define WS_W2 928
#define WS_C0 960
#define WS_U  1024

// ---------------------------------------------------------------- stage 0
__global__ __launch_bounds__(256)
void stage0_fuse(const float* __restrict__ WQ, const float* __restrict__ bQ,
                 const float* __restrict__ WK, const float* __restrict__ bK,
                 const float* __restrict__ WV, const float* __restrict__ bV,
                 const float* __restrict__ Ws, const float* __restrict__ bs,
                 const float* __restrict__ Wo, const float* __restrict__ bo,
                 float* __restrict__ ws) {
    const float inv = 0.316227766016838f; // 1/sqrt(10)
    int tid = threadIdx.x;
    // W1p : 32 rows x 28 cols, zero padded
    for (int i = tid; i < 32 * 28; i += blockDim.x) {
        int n = i / 28, k = i % 28;
        float acc = 0.f;
        if (n < 25 && k < 25) {
            #pragma unroll
            for (int d = 0; d < ATTN_DIM; ++d)
                acc = fmaf(WK[d * 25 + n], WQ[d * 25 + k], acc);
            acc *= inv;
        } else if (n == 25 && k < 25) {
            #pragma unroll
            for (int d = 0; d < ATTN_DIM; ++d)
                acc = fmaf(Ws[d * 25 + k], Wo[d], acc);
        }
        ws[WS_W1 + i] = acc;
    }
    // v1 (with zero sentinel at [25]), w2
    for (int t = tid; t < 26; t += blockDim.x) {
        float a = 0.f, b = 0.f;
        if (t < 25) {
            #pragma unroll
            for (int d = 0; d < ATTN_DIM; ++d) {
                a = fmaf(WK[d * 25 + t], bQ[d], a);
                b = fmaf(WV[d * 25 + t], Wo[d], b);
            }
            ws[WS_W2 + t] = b;
        }
        ws[WS_V1 + t] = a * inv;
    }
    if (tid == 0) {
        float c = bo[0];
        #pragma unroll
        for (int d = 0; d < ATTN_DIM; ++d)
            c = fmaf(Wo[d], bV[d] + bs[d], c);
        ws[WS_C0] = c;
    }
}

// ---------------------------------------------------------------- stage 1
// U[b, 0..25] = X_sub[b, :] @ W1p^T (+ v1 on cols 0..24) via v_wmma_f32_16x16x4_f32.
// One wave handles a 16-row b tile and one 16-col N tile; 7 K-steps of 4.
// B operand comes from the zero-padded 32x28 W1p -> all loads unconditional.
__global__ __launch_bounds__(256)
void stage1_wmma(const float* __restrict__ Xsub, const float* __restrict__ ws,
                 float* __restrict__ U) {
    int wid  = blockIdx.x * (blockDim.x >> 5) + (threadIdx.x >> 5);
    int lane = threadIdx.x & 31;
    int tile  = wid >> 1;          // b tile index [0, 1024)
    int ntile = wid & 1;           // 0: cols 0-15, 1: cols 16-31
    int b0 = tile * 16;

    int m  = lane & 15;            // A row (b within tile)
    int hi = lane >> 4;            // half-wave select
    int kb = hi * 2;               // K pair base within 4-wide K step
    int n  = ntile * 16 + m;       // output column / W1p row (always < 32)

    const float* __restrict__ W1 = ws + WS_W1;   // [32][28]
    const float* __restrict__ v1 = ws + WS_V1;
    const float* __restrict__ arow = Xsub + (size_t)(b0 + m) * 25;
    const float* __restrict__ brow = W1 + n * 28;

    v8f acc = {};
    // 6 full K-steps: ka+1 <= 23 < 25, all loads unconditional
    #pragma unroll
    for (int k0 = 0; k0 < 24; k0 += 4) {
        int ka = k0 + kb;
        v2f a, bmat;
        // A 16x4 layout: VGPR0 = K {0 | 2}, VGPR1 = K {1 | 3} (lanes 0-15 | 16-31)
        a.x = arow[ka];
        a.y = arow[ka + 1];
        // B 4x16 layout: row striped across lanes within a VGPR (K {0|2}, {1|3})
        bmat.x = brow[ka];
        bmat.y = brow[ka + 1];
        acc = __builtin_amdgcn_wmma_f32_16x16x4_f32(
            false, a, false, bmat, (short)0, acc, false, false);
    }
    // Peeled final K-step (K = 24..27): only K=24 is live on the A side.
    {
        float t24 = arow[24];              // unconditional load
        v2f a, bmat;
        a.x = (hi == 0) ? t24 : 0.f;       // v_cndmask, no branch
        a.y = 0.f;
        bmat.x = brow[24 + kb];            // pad region holds zeros
        bmat.y = brow[25 + kb];
        acc = __builtin_amdgcn_wmma_f32_16x16x4_f32(
            false, a, false, bmat, (short)0, acc, false, false);
    }
    // D 16x16 f32: VGPR j holds row (j + 8*hi), col = lane&15
    int col = ntile * 16 + m;
    if (col < 26) {
        float bias = v1[col];              // v1[25] == 0 sentinel
        #pragma unroll
        for (int j = 0; j < 8; ++j) {
            int row = b0 + j + hi * 8;
            U[(size_t)row * 32 + col] = acc[j] + bias;
        }
    }
}

// ---------------------------------------------------------------- stage 2
// One wave per b: stream x[b, r, :] once, online softmax, weighted-mean m,
// then out[b] = tanh(w2.m + h + C0) mapped to [-0.5, 3.41].
__global__ __launch_bounds__(256)
void stage2_stream(const float* __restrict__ Xrev, const float* __restrict__ ws,
                   float* __restrict__ out) {
    int b    = blockIdx.x * (blockDim.x >> 5) + (threadIdx.x >> 5);
    int lane = threadIdx.x & 31;
    if (b >= BATCH) return;

    const float* __restrict__ urow = ws + WS_U + (size_t)b * 32;
    float u[25];
    #pragma unroll
    for (int t = 0; t < 25; ++t) u[t] = urow[t];
    float h = urow[25];

    const float* __restrict__ xb = Xrev + (size_t)b * NREV * NUM_TOPICS;

    float lmax = -3.0e38f, lsum = 0.f;
    float macc[25];
    #pragma unroll
    for (int t = 0; t < 25; ++t) macc[t] = 0.f;

    for (int r = lane; r < NREV; r += 32) {
        const float* xr = xb + r * NUM_TOPICS;
        if (r + 32 < NREV)
            __builtin_prefetch(xr + 32 * NUM_TOPICS, 0, 1);
        float x[25];
        #pragma unroll
        for (int t = 0; t < 25; ++t) x[t] = xr[t];
        float s = 0.f;
        #pragma unroll
        for (int t = 0; t < 25; ++t) s = fmaf(x[t], u[t], s);
        // branch-free online softmax update
        float nm = fmaxf(lmax, s);
        float a  = __expf(lmax - nm);
        float w  = __expf(s - nm);
        lsum = fmaf(lsum, a, w);
        #pragma unroll
        for (int t = 0; t < 25; ++t) macc[t] = fmaf(macc[t], a, w * x[t]);
        lmax = nm;
    }

    // merge the 32 per-lane softmax states (max, sum, vec) via xor shuffles
    #pragma unroll
    for (int off = 16; off >= 1; off >>= 1) {
        float omax = __shfl_xor(lmax, off, 32);
        float osum = __shfl_xor(lsum, off, 32);
        float nm = fmaxf(lmax, omax);
        float a  = __expf(lmax - nm);
        float bb = __expf(omax - nm);
        lsum = lsum * a + osum * bb;
        #pragma unroll
        for (int t = 0; t < 25; ++t) {
            float om = __shfl_xor(macc[t], off, 32);
            macc[t] = macc[t] * a + om * bb;
        }
        lmax = nm;
    }

    if (lane == 0) {
        const float* w2 = ws + WS_W2;
        float y = 0.f;
        #pragma unroll
        for (int t = 0; t < 25; ++t) y = fmaf(w2[t], macc[t], y);
        y = y / lsum + h + ws[WS_C0];
        out[b] = fmaf(tanhf(y), SCALE_F, SCALE_F + TMIN_F);
    }
}

// ---------------------------------------------------------------- launcher
extern "C" void kernel_launch(void* const* d_in, const int* in_sizes, int n_in,
                              void* d_out, int out_size, void* d_ws, size_t ws_size,
                              hipStream_t stream) {
    const float* Xsub = (const float*)d_in[0];   // [B, 25]
    const float* Xrev = (const float*)d_in[1];   // [B, R, 25]
    const float* WQ = (const float*)d_in[2];
    const float* bQ = (const float*)d_in[3];
    const float* WK = (const float*)d_in[4];
    const float* bK = (const float*)d_in[5];   (void)bK; // dropped: softmax shift-invariant
    const float* WV = (const float*)d_in[6];
    const float* bV = (const float*)d_in[7];
    const float* Ws = (const float*)d_in[8];
    const float* bs = (const float*)d_in[9];
    const float* Wo = (const float*)d_in[10];
    const float* bo = (const float*)d_in[11];
    float* ws  = (float*)d_ws;
    float* U   = ws + WS_U;
    float* out = (float*)d_out;

    stage0_fuse<<<1, 256, 0, stream>>>(WQ, bQ, WK, bK, WV, bV, Ws, bs, Wo, bo, ws);

    // (B/16 tiles) * 2 N-tiles = 2048 waves; 8 waves/block
    stage1_wmma<<<256, 256, 0, stream>>>(Xsub, ws, U);

    // one wave per b: 16384 waves, 8/block
    stage2_stream<<<2048, 256, 0, stream>>>(Xrev, ws, out);
}